// AFP_Hetero_Molecular_82695300317253
// MI455X (gfx1250) — compile-verified
//
#include <hip/hip_runtime.h>
#include <math.h>

#define HH 4
#define CC 128
#define BG 256
#define TT 3

typedef __attribute__((ext_vector_type(2))) float v2f;
typedef __attribute__((ext_vector_type(8))) float v8f;

// ---------------------------------------------------------------------------
// Generic fp32 WMMA GEMM: D[M,N] = act( scale * (A[M,K] @ B[K,N]) + bias[n] )
// One wave (32 lanes) computes one 16x16 tile via V_WMMA_F32_16X16X4_F32.
// Layouts per CDNA5 ISA 7.12.2:
//   A 16x4 : lane l -> row m=l&15, element e -> k = 2*(l>>4)+e
//   B 4x16 : lane l -> col n=l&15, element e -> k = 2*(l>>4)+e
//   D 16x16: lane l, vgpr v -> m = v + 8*(l>>4), n = l&15
// ---------------------------------------------------------------------------
__global__ __launch_bounds__(32)
void k_gemm(const float* __restrict__ A, int lda,
            const float* __restrict__ B, int ldb,
            float* __restrict__ D, int ldd,
            int K, const float* __restrict__ bias,
            float scale, int act)
{
    const int lane = threadIdx.x & 31;
    const int half = lane >> 4;
    const int l15  = lane & 15;
    const int n0 = blockIdx.x * 16;
    const int m0 = blockIdx.y * 16;

    v8f acc = {0.f,0.f,0.f,0.f,0.f,0.f,0.f,0.f};
    const float* __restrict__ Arow = A + (long)(m0 + l15) * lda;
    const float* __restrict__ Bcol = B + (n0 + l15);

    for (int k0 = 0; k0 < K; k0 += 4) {
        const int ka = k0 + half * 2;
        v2f a, b;
        a.x = Arow[ka];
        a.y = Arow[ka + 1];
        b.x = Bcol[(long)ka * ldb];
        b.y = Bcol[(long)(ka + 1) * ldb];
        acc = __builtin_amdgcn_wmma_f32_16x16x4_f32(
                false, a, false, b, (short)0, acc, false, false);
    }

    const int n = n0 + l15;
    const float bv = bias ? bias[n] : 0.f;
#pragma unroll
    for (int v = 0; v < 8; ++v) {
        const int m = m0 + v + half * 8;
        float val = acc[v] * scale + bv;
        if (act == 1) val = (val > 0.f) ? val : (__expf(val) - 1.f);  // ELU
        D[(long)m * ldd + n] = val;
    }
}

// V[f][h] = sum_c W[f, h*128+c] * att[h, c]   (folds att into projection)
__global__ void k_vhead(const float* __restrict__ W,
                        const float* __restrict__ att,
                        float* __restrict__ V)
{
    int t = blockIdx.x * blockDim.x + threadIdx.x;   // 512 threads
    if (t >= CC * HH) return;
    int f = t >> 2, h = t & 3;
    float s = 0.f;
    for (int c = 0; c < CC; ++c)
        s += W[(long)f * (HH * CC) + h * CC + c] * att[h * CC + c];
    V[f * HH + h] = s;
}

// Bproj[(h*128+f)][c] = Wsrc[f][h*128+c]  : permute Wsrc into GEMM-B layout
__global__ void k_bproj(const float* __restrict__ Wsrc, float* __restrict__ Bp)
{
    int t = blockIdx.x * blockDim.x + threadIdx.x;   // 512*128 threads
    if (t >= HH * CC * CC) return;
    int row = t >> 7;           // h*128+f
    int c   = t & 127;
    int h = row >> 7, f = row & 127;
    Bp[t] = Wsrc[(long)f * (HH * CC) + h * CC + c];
}

// out[c*rows + r] = in[r*cols + c]
__global__ void k_transpose(const float* __restrict__ in, int rows, int cols,
                            float* __restrict__ out)
{
    int t = blockIdx.x * blockDim.x + threadIdx.x;
    if (t >= rows * cols) return;
    int r = t / cols, c = t % cols;
    out[(long)c * rows + r] = in[t];
}

__global__ void k_copy(const float* __restrict__ in, float* __restrict__ out, int n)
{
    int t = blockIdx.x * blockDim.x + threadIdx.x;
    if (t < n) out[t] = in[t];
}

// dst is sorted; build CSR offsets[g] = first edge with dst >= g, offsets[B]=E
__global__ void k_offsets(const int* __restrict__ dst, int E, int nb,
                          int* __restrict__ offs)
{
    int i = blockIdx.x * blockDim.x + threadIdx.x;
    if (i >= E) return;
    int d = dst[i];
    int p = (i == 0) ? -1 : dst[i - 1];
    for (int g = p + 1; g <= d; ++g) offs[g] = i;
    if (i == E - 1)
        for (int g = d + 1; g <= nb; ++g) offs[g] = E;
}

// out[n][h] = X[n,:] . V[:,h]  (one wave32 per row, shfl reduction)
__global__ __launch_bounds__(256)
void k_rowdot4(const float* __restrict__ X, const float* __restrict__ V,
               float* __restrict__ out, int N)
{
    int w = (blockIdx.x * 256 + threadIdx.x) >> 5;
    int lane = threadIdx.x & 31;
    if (w >= N) return;
    const float* __restrict__ row = X + (long)w * CC;
    float a0 = 0.f, a1 = 0.f, a2 = 0.f, a3 = 0.f;
    for (int f = lane; f < CC; f += 32) {
        float v = row[f];
        a0 += v * V[f * 4 + 0];
        a1 += v * V[f * 4 + 1];
        a2 += v * V[f * 4 + 2];
        a3 += v * V[f * 4 + 3];
    }
#pragma unroll
    for (int off = 16; off > 0; off >>= 1) {
        a0 += __shfl_down(a0, off, 32);
        a1 += __shfl_down(a1, off, 32);
        a2 += __shfl_down(a2, off, 32);
        a3 += __shfl_down(a3, off, 32);
    }
    if (lane == 0) {
        out[w * 4 + 0] = a0; out[w * 4 + 1] = a1;
        out[w * 4 + 2] = a2; out[w * 4 + 3] = a3;
    }
}

// Per-graph segment softmax (max + exp + sum), dst-contiguous. Deterministic.
__global__ __launch_bounds__(256)
void k_softmax(const float* __restrict__ a_s, const float* __restrict__ a_d,
               const int* __restrict__ src, const int* __restrict__ offs,
               float* __restrict__ ex, float* __restrict__ den)
{
    __shared__ float red[256];
    const int g = blockIdx.x;
    const int s0 = offs[g], s1 = offs[g + 1];
    const int tid = threadIdx.x;
    float ad[4], mx[4];
#pragma unroll
    for (int h = 0; h < 4; ++h) { ad[h] = a_d[g * 4 + h]; mx[h] = -3.0e38f; }

    for (int e = s0 + tid; e < s1; e += 256) {
        int n = src[e];
#pragma unroll
        for (int h = 0; h < 4; ++h) {
            float al = a_s[n * 4 + h] + ad[h];
            al = (al > 0.f) ? al : 0.01f * al;          // leaky_relu
            mx[h] = fmaxf(mx[h], al);
        }
    }
#pragma unroll
    for (int h = 0; h < 4; ++h) {
        red[tid] = mx[h]; __syncthreads();
        for (int s = 128; s > 0; s >>= 1) {
            if (tid < s) red[tid] = fmaxf(red[tid], red[tid + s]);
            __syncthreads();
        }
        mx[h] = red[0]; __syncthreads();
    }

    float sm[4] = {0.f, 0.f, 0.f, 0.f};
    for (int e = s0 + tid; e < s1; e += 256) {
        int n = src[e];
#pragma unroll
        for (int h = 0; h < 4; ++h) {
            float al = a_s[n * 4 + h] + ad[h];
            al = (al > 0.f) ? al : 0.01f * al;
            float xv = __expf(al - mx[h]);
            ex[e * 4 + h] = xv;
            sm[h] += xv;
        }
    }
#pragma unroll
    for (int h = 0; h < 4; ++h) {
        red[tid] = sm[h]; __syncthreads();
        for (int s = 128; s > 0; s >>= 1) {
            if (tid < s) red[tid] += red[tid + s];
            __syncthreads();
        }
        if (tid == 0) den[g * 4 + h] = red[0];
        __syncthreads();
    }
}

// z[g][h*128+f] = ( sum_{e in g} ex[e,h] * x[src_e, f] ) / den[g,h]
__global__ __launch_bounds__(128)
void k_zagg(const float* __restrict__ x, const int* __restrict__ src,
            const int* __restrict__ offs, const float* __restrict__ ex,
            const float* __restrict__ den, float* __restrict__ z)
{
    const int g = blockIdx.x;
    const int f = threadIdx.x;
    const int s0 = offs[g], s1 = offs[g + 1];
    float acc[4] = {0.f, 0.f, 0.f, 0.f};
    for (int e = s0; e < s1; ++e) {
        int n = src[e];
        float xv = x[(long)n * CC + f];
#pragma unroll
        for (int h = 0; h < 4; ++h) acc[h] += ex[e * 4 + h] * xv;
    }
#pragma unroll
    for (int h = 0; h < 4; ++h) {
        float d = den[g * 4 + h];
        float inv = (d > 0.f) ? (1.f / d) : 0.f;        // empty segment -> 0
        z[(long)g * (HH * CC) + h * CC + f] = acc[h] * inv;
    }
}

// GRUCell gates + ReLU, emb updated in place
__global__ void k_gru(const float* __restrict__ gi, const float* __restrict__ gh,
                      float* __restrict__ emb)
{
    int t = blockIdx.x * blockDim.x + threadIdx.x;     // BG*CC
    if (t >= BG * CC) return;
    int g = t >> 7, c = t & 127;
    const float* gir = gi + (long)g * 3 * CC;
    const float* ghr = gh + (long)g * 3 * CC;
    float r = 1.f / (1.f + __expf(-(gir[c]        + ghr[c])));
    float zz = 1.f / (1.f + __expf(-(gir[CC + c]  + ghr[CC + c])));
    float n  = tanhf(gir[2 * CC + c] + r * ghr[2 * CC + c]);
    float e  = (1.f - zz) * n + zz * emb[t];
    emb[t] = (e > 0.f) ? e : 0.f;                       // ReLU
}

// ---------------------------------------------------------------------------
static inline char* bump(char*& p, size_t bytes)
{
    char* r = p;
    p += (bytes + 255) & ~(size_t)255;
    return r;
}

static void run_branch(const float* x, const float* emb_in,
                       const int* src, const int* dst,
                       void* const* prm, int N, float* y,
                       char*& cur, hipStream_t stream)
{
    const float* Wsrc = (const float*)prm[0];
    const float* Wdst = (const float*)prm[1];
    const float* atts = (const float*)prm[2];
    const float* attd = (const float*)prm[3];
    const float* bias = (const float*)prm[4];
    const float* Wih  = (const float*)prm[5];
    const float* Whh  = (const float*)prm[6];
    const float* bih  = (const float*)prm[7];
    const float* bhh  = (const float*)prm[8];
    const float* linW = (const float*)prm[9];
    const float* linb = (const float*)prm[10];

    float* Vs    = (float*)bump(cur, 512 * 4);
    float* Vd    = (float*)bump(cur, 512 * 4);
    float* Bp    = (float*)bump(cur, (size_t)512 * 128 * 4);
    float* WihT  = (float*)bump(cur, (size_t)128 * 384 * 4);
    float* WhhT  = (float*)bump(cur, (size_t)128 * 384 * 4);
    float* linWT = (float*)bump(cur, (size_t)128 * 128 * 4);
    int*   offs  = (int*)  bump(cur, (BG + 1) * 4);
    float* a_s   = (float*)bump(cur, (size_t)N * 4 * 4);
    float* a_d   = (float*)bump(cur, BG * 4 * 4);
    float* exw   = (float*)bump(cur, (size_t)N * 4 * 4);
    float* den   = (float*)bump(cur, BG * 4 * 4);
    float* z     = (float*)bump(cur, (size_t)BG * 512 * 4);
    float* emb   = (float*)bump(cur, (size_t)BG * 128 * 4);
    float* hbuf  = (float*)bump(cur, (size_t)BG * 128 * 4);
    float* gi    = (float*)bump(cur, (size_t)BG * 384 * 4);
    float* gh    = (float*)bump(cur, (size_t)BG * 384 * 4);

    // ---- prep (cheap, recomputed every call: deterministic, no caching) ----
    k_vhead<<<2, 256, 0, stream>>>(Wsrc, atts, Vs);
    k_vhead<<<2, 256, 0, stream>>>(Wdst, attd, Vd);
    k_bproj<<<(512 * 128 + 255) / 256, 256, 0, stream>>>(Wsrc, Bp);
    k_transpose<<<(384 * 128 + 255) / 256, 256, 0, stream>>>(Wih, 384, 128, WihT);
    k_transpose<<<(384 * 128 + 255) / 256, 256, 0, stream>>>(Whh, 384, 128, WhhT);
    k_transpose<<<(128 * 128 + 255) / 256, 256, 0, stream>>>(linW, 128, 128, linWT);
    k_copy<<<(BG * 128 + 255) / 256, 256, 0, stream>>>(emb_in, emb, BG * 128);
    k_offsets<<<(N + 255) / 256, 256, 0, stream>>>(dst, N, BG, offs);
    k_rowdot4<<<((size_t)N * 32 + 255) / 256, 256, 0, stream>>>(x, Vs, a_s, N);

    // ---- T timesteps ----
    for (int t = 0; t < TT; ++t) {
        k_rowdot4<<<(BG * 32 + 255) / 256, 256, 0, stream>>>(emb, Vd, a_d, BG);
        k_softmax<<<BG, 256, 0, stream>>>(a_s, a_d, src, offs, exw, den);
        k_zagg<<<BG, 128, 0, stream>>>(x, src, offs, exw, den, z);
        // h = elu( mean_h( (z/den) @ Wsrc_h ) + bias )
        k_gemm<<<dim3(128 / 16, BG / 16), 32, 0, stream>>>(
            z, 512, Bp, 128, hbuf, 128, 512, bias, 0.25f, 1);
        // gi = h @ Wih^T + bih ; gh = emb @ Whh^T + bhh
        k_gemm<<<dim3(384 / 16, BG / 16), 32, 0, stream>>>(
            hbuf, 128, WihT, 384, gi, 384, 128, bih, 1.f, 0);
        k_gemm<<<dim3(384 / 16, BG / 16), 32, 0, stream>>>(
            emb, 128, WhhT, 384, gh, 384, 128, bhh, 1.f, 0);
        k_gru<<<(BG * 128 + 255) / 256, 256, 0, stream>>>(gi, gh, emb);
    }

    // y = emb @ linW^T + linb
    k_gemm<<<dim3(128 / 16, BG / 16), 32, 0, stream>>>(
        emb, 128, linWT, 128, y, 128, 128, linb, 1.f, 0);
}

extern "C" void kernel_launch(void* const* d_in, const int* in_sizes, int n_in,
                              void* d_out, int out_size, void* d_ws, size_t ws_size,
                              hipStream_t stream)
{
    const float* x_pa   = (const float*)d_in[0];
    const float* emb_pa = (const float*)d_in[1];
    const float* x_la   = (const float*)d_in[2];
    const float* emb_la = (const float*)d_in[3];
    const int*   src_pa = (const int*)d_in[4];
    const int*   dst_pa = (const int*)d_in[5];
    const int*   src_la = (const int*)d_in[6];
    const int*   dst_la = (const int*)d_in[7];
    const int N_pa = in_sizes[0] / CC;
    const int N_la = in_sizes[2] / CC;

    float* out = (float*)d_out;
    char* cur = (char*)d_ws;

    run_branch(x_pa, emb_pa, src_pa, dst_pa, &d_in[8],  N_pa, out,            cur, stream);
    run_branch(x_la, emb_la, src_la, dst_la, &d_in[19], N_la, out + BG * CC,  cur, stream);
}